// _DifferentiableNeuralDictionary_71734543778598
// MI455X (gfx1250) — compile-verified
//
#include <hip/hip_runtime.h>

typedef __attribute__((ext_vector_type(2))) float v2f;
typedef __attribute__((ext_vector_type(4))) float v4f;
typedef __attribute__((ext_vector_type(8))) float v8f;

#define DND_DELTA 1e-3f

// Problem constants (from the reference setup)
constexpr int DND_B = 4096;
constexpr int DND_D = 512;
constexpr int DND_K = 50;

constexpr int WAVES = 16;            // waves per block (wave32)
constexpr int TPB   = WAVES * 32;    // 512 threads

__global__ __launch_bounds__(TPB)
void dnd_wmma_kernel(const float* __restrict__ key,          // [B, D]
                     const float* __restrict__ keys_table,   // [N, D]
                     const float* __restrict__ values_table, // [N]
                     const int*   __restrict__ indices,      // [B, K]
                     float*       __restrict__ out)          // [B]
{
    // Query tile staged in LDS: x reads go through the DS pipe so the VMEM
    // pipe is dedicated to the (bandwidth-bound) gathered y rows.
    __shared__ float s_x[16 * DND_D];            // 32 KB
    __shared__ float s_num[WAVES][16];
    __shared__ float s_den[WAVES][16];

    const int tile = blockIdx.x;                // 16 batch rows per block
    const int lane = threadIdx.x & 31;
    const int wave = threadIdx.x >> 5;
    const int j    = lane & 15;                 // row-in-tile owned by this lane
    const int off  = (lane >> 4) * 2;           // WMMA fragment column sub-offset {0,2}
    const int brow = tile * 16 + j;

    // Stage the contiguous 32 KB query tile into LDS (coalesced b128 copy).
    {
        const v4f* src = (const v4f*)(key + (size_t)tile * 16 * DND_D);
        v4f*       dst = (v4f*)s_x;
        #pragma unroll
        for (int i = threadIdx.x; i < (16 * DND_D) / 4; i += TPB)
            dst[i] = src[i];
    }
    __syncthreads();

    const float* xl = s_x + j * DND_D;          // this lane's query row in LDS

    // Diagonal element (j,j) of the 16x16 f32 accumulator lives in:
    //   lane j,     VGPR j     (j <  8)
    //   lane j+16,  VGPR j-8   (j >= 8)
    const bool holder = (lane < 8) || (lane >= 24);
    const int  r      = (lane < 16) ? j : (j - 8);   // acc register index (holders only)

    float num = 0.0f;
    float den = 0.0f;

    for (int k = wave; k < DND_K; k += WAVES) {
        const int idx = indices[(size_t)brow * DND_K + k];
        const float* yrow = keys_table + (size_t)idx * DND_D;

        v8f acc0 = {};
        v8f acc1 = {};
        #pragma unroll 4
        for (int c = 0; c < DND_D; c += 8) {
            v2f x0 = *(const v2f*)(xl + c + off);          // ds_load_b64
            v2f y0 = *(const v2f*)(yrow + c + off);        // global_load_b64
            v2f x1 = *(const v2f*)(xl + c + 4 + off);
            v2f y1 = *(const v2f*)(yrow + c + 4 + off);
            v2f d0 = x0 - y0;
            v2f d1 = x1 - y1;
            // C[i][j] += sum_k (x_i - y_i)[k] * (x_j - y_j)[k]; diagonal = dist^2
            acc0 = __builtin_amdgcn_wmma_f32_16x16x4_f32(
                       false, d0, false, d0, (short)0, acc0, false, false);
            acc1 = __builtin_amdgcn_wmma_f32_16x16x4_f32(
                       false, d1, false, d1, (short)0, acc1, false, false);
        }
        v8f acc = acc0 + acc1;

        // Extract this lane's diagonal element (meaningful for holder lanes only).
        float d2 = acc[0];
        if (r == 1) d2 = acc[1];
        if (r == 2) d2 = acc[2];
        if (r == 3) d2 = acc[3];
        if (r == 4) d2 = acc[4];
        if (r == 5) d2 = acc[5];
        if (r == 6) d2 = acc[6];
        if (r == 7) d2 = acc[7];

        if (holder) {
            float w = 1.0f / (d2 + DND_DELTA);
            float v = values_table[idx];
            num += w * v;
            den += w;
        }
    }

    // Each row j has exactly one holder lane per wave: deterministic LDS slots.
    if (holder) {
        s_num[wave][j] = num;
        s_den[wave][j] = den;
    }
    __syncthreads();

    if (threadIdx.x < 16) {
        float n = 0.0f, d = 0.0f;
        #pragma unroll
        for (int w = 0; w < WAVES; ++w) {
            n += s_num[w][threadIdx.x];
            d += s_den[w][threadIdx.x];
        }
        out[tile * 16 + threadIdx.x] = n / d;
    }
}

extern "C" void kernel_launch(void* const* d_in, const int* in_sizes, int n_in,
                              void* d_out, int out_size, void* d_ws, size_t ws_size,
                              hipStream_t stream) {
    const float* key          = (const float*)d_in[0];   // [4096, 512]
    const float* keys_table   = (const float*)d_in[1];   // [100000, 512]
    const float* values_table = (const float*)d_in[2];   // [100000]
    const int*   indices      = (const int*)d_in[3];     // [4096, 50]
    float*       out          = (float*)d_out;           // [4096]

    (void)in_sizes; (void)n_in; (void)out_size; (void)d_ws; (void)ws_size;

    dim3 grid(DND_B / 16);
    dim3 block(TPB);
    dnd_wmma_kernel<<<grid, block, 0, stream>>>(key, keys_table, values_table,
                                                indices, out);
}